// Attentionv2_37349035606151
// MI455X (gfx1250) — compile-verified
//
#include <hip/hip_runtime.h>
#include <hip/hip_bf16.h>
#include <math.h>

#define FEATURE 512
#define HIDDEN  512
#define ATTN    256
#define S_DIM   512
#define B_DIM   8
#define T_DIM   128

typedef float v2f __attribute__((ext_vector_type(2)));
typedef float v8f __attribute__((ext_vector_type(8)));

// gfx1250 native transcendental tanh (V_TANH_F32).
__device__ __forceinline__ float dev_tanh(float x) {
#if __has_builtin(__builtin_amdgcn_tanhf)
    return __builtin_amdgcn_tanhf(x);
#elif __has_builtin(__builtin_amdgcn_tanh_f32)
    return __builtin_amdgcn_tanh_f32(x);
#else
    float r;
    // TRANS op: need 1 independent op (V_NOP) before result use (ISA 7.4#3).
    asm volatile("v_tanh_f32 %0, %1\n\tv_nop" : "=v"(r) : "v"(x));
    return r;
#endif
}

// ---------------------------------------------------------------------------
// Kernel 1/2: out[r, 0:256] = X[r, 0:512] @ W[256,512]^T + bias   (fp32 WMMA)
// One wave = one 16x16 output tile, V_WMMA_F32_16X16X4_F32, K-loop of 128
// with uniform (lane-independent) trip count so the compiler can unroll and
// pipeline loads ahead of the WMMA pipe; the per-lane K-pair offset (2*half)
// is folded into the base pointers.
// A layout: lane%16 = M row, lane/16 selects K pair {0,1} vs {2,3}.
// B layout mirrors with N = lane%16 (B[k,n] = W[n,k]).
// ---------------------------------------------------------------------------
__global__ __launch_bounds__(128) void proj_gemm_kernel(
    const float* __restrict__ X,     // [R, 512]
    const float* __restrict__ W,     // [256, 512]
    const float* __restrict__ bias,  // [256]
    float* __restrict__ out)         // [R, 256]
{
    const int wave = threadIdx.x >> 5;
    const int lane = threadIdx.x & 31;
    const int half = lane >> 4;
    const int mn   = lane & 15;

    const int tile = blockIdx.x * 4 + wave;   // grid sized exactly
    const int rt   = tile >> 4;               // row tile (16 rows)
    const int ct   = tile & 15;               // col tile (16 of 256 cols)

    // fold the per-half K offset into the pointers -> uniform loop bounds
    const float* Arow = X + (size_t)(rt * 16 + mn) * 512 + 2 * half;
    const float* Brow = W + (size_t)(ct * 16 + mn) * 512 + 2 * half;

    v8f c = {};
    #pragma unroll 8
    for (int k0 = 0; k0 < 512; k0 += 4) {
        v2f a = *(const v2f*)(Arow + k0);
        v2f b = *(const v2f*)(Brow + k0);
        c = __builtin_amdgcn_wmma_f32_16x16x4_f32(
                false, a, false, b, (short)0, c, false, false);
    }

    const float bv = bias[ct * 16 + mn];
    #pragma unroll
    for (int v = 0; v < 8; ++v) {
        const int m = v + 8 * half;           // C/D layout: M = v + 8*half
        out[(size_t)(rt * 16 + m) * 256 + ct * 16 + mn] = c[v] + bv;
    }
}

// ---------------------------------------------------------------------------
// Kernel 3: fused scores + mask + softmax.
// One block per (b,t); 256 threads, each owns scores for s = tid and tid+256.
// score[s] = vb + sum_a va[a] * tanh(proj_img[s,b,a] + proj_hid[t,b,a])
// Never materializes the [B,T,S,A] tensor. Bound by V_TANH_F32 throughput.
// ---------------------------------------------------------------------------
__global__ __launch_bounds__(256) void score_softmax_kernel(
    const float* __restrict__ proj_img,   // [S, B, 256]
    const float* __restrict__ proj_hid,   // [T, B, 256]
    const float* __restrict__ va_w,       // [256]
    const float* __restrict__ va_b,       // [1]
    const unsigned char* __restrict__ mask, // [B, T, S] (bool)
    float* __restrict__ wgt)              // [T, B, S]
{
    const int b = blockIdx.x & (B_DIM - 1);
    const int t = blockIdx.x >> 3;
    const int tid = threadIdx.x;

    __shared__ float hidS[ATTN];
    __shared__ float vaS[ATTN];
    __shared__ float red[256];

    hidS[tid] = proj_hid[(size_t)(t * B_DIM + b) * ATTN + tid];
    vaS[tid]  = va_w[tid];
    __syncthreads();

    const float vb = va_b[0];
    float sc[2];
    #pragma unroll
    for (int i = 0; i < 2; ++i) {
        const int s = tid + i * 256;
        const float* ip = proj_img + (size_t)(s * B_DIM + b) * ATTN;
        float acc = 0.0f;
        #pragma unroll 4
        for (int a = 0; a < ATTN; a += 4) {
            const float4 iv = *(const float4*)(ip + a);
            acc += vaS[a + 0] * dev_tanh(iv.x + hidS[a + 0]);
            acc += vaS[a + 1] * dev_tanh(iv.y + hidS[a + 1]);
            acc += vaS[a + 2] * dev_tanh(iv.z + hidS[a + 2]);
            acc += vaS[a + 3] * dev_tanh(iv.w + hidS[a + 3]);
        }
        acc += vb;
        if (!mask[(size_t)(b * T_DIM + t) * S_DIM + s]) acc = -INFINITY;
        sc[i] = acc;
    }

    // block max over 512 scores
    red[tid] = fmaxf(sc[0], sc[1]);
    __syncthreads();
    for (int off = 128; off > 0; off >>= 1) {
        if (tid < off) red[tid] = fmaxf(red[tid], red[tid + off]);
        __syncthreads();
    }
    const float mx = red[0];
    __syncthreads();

    const float e0 = __expf(sc[0] - mx);
    const float e1 = __expf(sc[1] - mx);
    red[tid] = e0 + e1;
    __syncthreads();
    for (int off = 128; off > 0; off >>= 1) {
        if (tid < off) red[tid] += red[tid + off];
        __syncthreads();
    }
    const float inv = 1.0f / red[0];

    float* wrow = wgt + (size_t)(t * B_DIM + b) * S_DIM;
    wrow[tid]       = e0 * inv;
    wrow[tid + 256] = e1 * inv;
}

// ---------------------------------------------------------------------------
// Kernel 4: context[t,b,f] = sum_s w[t,b,s] * image_features[s,b,f]  (WMMA)
// Per b: GEMM [T=128 x S=512] x [S=512 x F=512]. One wave per 16x16 tile,
// uniform K-loop (per-half offset folded into pointers).
// ---------------------------------------------------------------------------
__global__ __launch_bounds__(128) void context_gemm_kernel(
    const float* __restrict__ wgt,   // [T, B, S]
    const float* __restrict__ img,   // [S, B, 512]
    float* __restrict__ ctx)         // [T, B, 512]
{
    const int wave = threadIdx.x >> 5;
    const int lane = threadIdx.x & 31;
    const int half = lane >> 4;
    const int mn   = lane & 15;

    const int tile = blockIdx.x * 4 + wave;   // 2048 tiles total
    const int b    = tile >> 8;               // 256 tiles per batch
    const int rem  = tile & 255;
    const int tt   = rem >> 5;                // 0..7  (T tiles)
    const int ft   = rem & 31;                // 0..31 (F tiles)

    const float* Arow = wgt + (size_t)((tt * 16 + mn) * B_DIM + b) * S_DIM
                            + 2 * half;
    const float* Bcol = img + (size_t)b * FEATURE + ft * 16 + mn
                            + (size_t)(2 * half) * (B_DIM * FEATURE);

    v8f c = {};
    #pragma unroll 8
    for (int k0 = 0; k0 < S_DIM; k0 += 4) {
        v2f a = *(const v2f*)(Arow + k0);     // contiguous in s
        v2f bb;
        bb.x = Bcol[(size_t)k0       * (B_DIM * FEATURE)];
        bb.y = Bcol[(size_t)(k0 + 1) * (B_DIM * FEATURE)];
        c = __builtin_amdgcn_wmma_f32_16x16x4_f32(
                false, a, false, bb, (short)0, c, false, false);
    }

    #pragma unroll
    for (int v = 0; v < 8; ++v) {
        const int m = v + 8 * half;
        ctx[(size_t)((tt * 16 + m) * B_DIM + b) * FEATURE + ft * 16 + mn] = c[v];
    }
}

extern "C" void kernel_launch(void* const* d_in, const int* in_sizes, int n_in,
                              void* d_out, int out_size, void* d_ws, size_t ws_size,
                              hipStream_t stream) {
    (void)in_sizes; (void)n_in; (void)out_size; (void)ws_size;

    const float* last_hiddens   = (const float*)d_in[0];  // [T,B,512]
    const float* image_features = (const float*)d_in[1];  // [S,B,512]
    const unsigned char* mask   = (const unsigned char*)d_in[2]; // [B,T,S] bool
    const float* Wa_w = (const float*)d_in[3];            // [256,512]
    const float* Wa_b = (const float*)d_in[4];            // [256]
    const float* Ua_w = (const float*)d_in[5];            // [256,512]
    const float* Ua_b = (const float*)d_in[6];            // [256]
    const float* va_w = (const float*)d_in[7];            // [256]
    const float* va_b = (const float*)d_in[8];            // [1]

    float* ctx = (float*)d_out;                                   // [T,B,512]
    float* wgt = (float*)d_out + (size_t)T_DIM * B_DIM * FEATURE; // [T,B,S]

    float* proj_img = (float*)d_ws;                               // [S,B,256]
    float* proj_hid = proj_img + (size_t)S_DIM * B_DIM * ATTN;    // [T,B,256]

    // proj_img: 4096x256 tiles -> 4096 waves -> 1024 blocks of 4 waves
    proj_gemm_kernel<<<1024, 128, 0, stream>>>(image_features, Wa_w, Wa_b, proj_img);
    // proj_hid: 1024x256 tiles -> 1024 waves -> 256 blocks
    proj_gemm_kernel<<<256, 128, 0, stream>>>(last_hiddens, Ua_w, Ua_b, proj_hid);
    // fused tanh-score + softmax: one block per (b,t)
    score_softmax_kernel<<<B_DIM * T_DIM, 256, 0, stream>>>(
        proj_img, proj_hid, va_w, va_b, mask, wgt);
    // context: 2048 tiles -> 512 blocks
    context_gemm_kernel<<<512, 128, 0, stream>>>(wgt, image_features, ctx);
}